// RelationCommLayer_18683107737873
// MI455X (gfx1250) — compile-verified
//
#include <hip/hip_runtime.h>
#include <hip/hip_bf16.h>
#include <math.h>

typedef __attribute__((ext_vector_type(2))) float v2f;
typedef __attribute__((ext_vector_type(8))) float v8f;

#define Bsz   8
#define Aseq  1024
#define Dm    256
#define Hh    4
#define HDd   64
#define Rr    5
#define MrowsT (Bsz*Aseq)

// D = A(16x4) * B(4x16) + C, fp32 WMMA (wave32).
__device__ __forceinline__ v8f wmma4f32(v2f a, v2f b, v8f c) {
  return __builtin_amdgcn_wmma_f32_16x16x4_f32(false, a, false, b, (short)0, c,
                                               false, false);
}

// ---------------------------------------------------------------------------
// Kernel 1: fused QKV projection.  x[8192x256] @ {Wq,Wk,Wv}[256x256] + bias
// -> q/k/v in [B,H,A,HD] layout.  Block = 4 waves sharing a 32x256 A tile in
// LDS; each wave computes 2 m-tiles per Wq/Wk/Wv fragment load (6 WMMA per
// k-step off one load set) to halve weight-matrix L2 traffic.
// ---------------------------------------------------------------------------
__global__ __launch_bounds__(128) void qkv_gemm_kernel(
    const float* __restrict__ x,
    const float* __restrict__ Wq, const float* __restrict__ bq,
    const float* __restrict__ Wk, const float* __restrict__ bk,
    const float* __restrict__ Wv, const float* __restrict__ bv,
    float* __restrict__ q, float* __restrict__ k, float* __restrict__ v) {
  __shared__ float ldsA[32 * 264];
  const int tid = threadIdx.x;
  const int m0 = blockIdx.x * 32;
  for (int u = tid; u < 32 * 64; u += 128) {
    const int row = u >> 6, c4 = u & 63;
    const float4 val =
        reinterpret_cast<const float4*>(x + (size_t)(m0 + row) * Dm)[c4];
    reinterpret_cast<float4*>(&ldsA[row * 264])[c4] = val;
  }
  __syncthreads();
  const int lane = tid & 31, wv = tid >> 5;
  const int nl = lane & 15, half = lane >> 4, kb = half * 2;
  const int n = (blockIdx.y * 4 + wv) * 16 + nl;
  const v8f vzero = {};
  v8f aq[2] = {vzero, vzero}, ak[2] = {vzero, vzero}, av[2] = {vzero, vzero};
#pragma unroll 2
  for (int s = 0; s < 64; ++s) {
    const int k0 = 4 * s + kb;
    v2f a0, a1, bq2, bk2, bv2;
    a0.x = ldsA[nl * 264 + k0];        a0.y = ldsA[nl * 264 + k0 + 1];
    a1.x = ldsA[(16 + nl) * 264 + k0]; a1.y = ldsA[(16 + nl) * 264 + k0 + 1];
    bq2.x = Wq[k0 * Dm + n]; bq2.y = Wq[(k0 + 1) * Dm + n];
    bk2.x = Wk[k0 * Dm + n]; bk2.y = Wk[(k0 + 1) * Dm + n];
    bv2.x = Wv[k0 * Dm + n]; bv2.y = Wv[(k0 + 1) * Dm + n];
    aq[0] = wmma4f32(a0, bq2, aq[0]);  aq[1] = wmma4f32(a1, bq2, aq[1]);
    ak[0] = wmma4f32(a0, bk2, ak[0]);  ak[1] = wmma4f32(a1, bk2, ak[1]);
    av[0] = wmma4f32(a0, bv2, av[0]);  av[1] = wmma4f32(a1, bv2, av[1]);
  }
  const int h = n >> 6, hd = n & 63;
  const float bqv = bq[n], bkv = bk[n], bvv = bv[n];
#pragma unroll
  for (int mt = 0; mt < 2; ++mt) {
#pragma unroll
    for (int r = 0; r < 8; ++r) {
      const int g = m0 + mt * 16 + r + half * 8;
      const int bidx = g >> 10, aidx = g & 1023;
      const size_t dst = ((size_t)(bidx * Hh + h) * Aseq + aidx) * HDd + hd;
      q[dst] = aq[mt][r] + bqv;
      k[dst] = ak[mt][r] + bkv;
      v[dst] = av[mt][r] + bvv;
    }
  }
}

// ---------------------------------------------------------------------------
// Kernel 2: fused attention.  Block = 4 waves, one (b,h), four consecutive
// 16-row i-tiles.  Each j-tile: K/V staged once into LDS for all 4 waves,
// S = Q Kt (WMMA), rel bias from the streamed rel_feat inline,
// p = adj ? exp(s)*w : 0  (softmax Z cancels against edge-weight renorm),
// P transposed C->A layout through per-wave LDS scratch, O += P V (WMMA),
// row-normalized at the end.
// ---------------------------------------------------------------------------
__global__ __launch_bounds__(128) void attn_kernel(
    const float* __restrict__ q, const float* __restrict__ k,
    const float* __restrict__ v, const float* __restrict__ rel,
    const unsigned char* __restrict__ adj, const float* __restrict__ ew,
    const float* __restrict__ Wr, const float* __restrict__ br,
    float* __restrict__ attn_out) {
  __shared__ float ldsK[16 * 68];
  __shared__ float ldsV[16 * 68];
  __shared__ float ldsP[4 * 16 * 17];
  const int tid = threadIdx.x;
  const int lane = tid & 31, wv = tid >> 5;
  const int i0 = blockIdx.x * 64 + wv * 16;
  const int h = blockIdx.y;
  const int b = blockIdx.z;
  const int bh = b * Hh + h;
  const int nl = lane & 15, half = lane >> 4, kb = half * 2;
  float* myP = &ldsP[wv * 16 * 17];
  const float wr0 = Wr[0 * Hh + h], wr1 = Wr[1 * Hh + h], wr2 = Wr[2 * Hh + h],
              wr3 = Wr[3 * Hh + h], wr4 = Wr[4 * Hh + h], brh = br[h];
  v2f qa[16];
  {
    const float* qrow = q + ((size_t)bh * Aseq + i0 + nl) * HDd;
#pragma unroll
    for (int s = 0; s < 16; ++s) {
      qa[s].x = qrow[4 * s + kb];
      qa[s].y = qrow[4 * s + kb + 1];
    }
  }
  const float* kbase = k + (size_t)bh * Aseq * HDd;
  const float* vbase = v + (size_t)bh * Aseq * HDd;
  v8f o0 = {}, o1 = {}, o2 = {}, o3 = {};
  float lsum[8];
#pragma unroll
  for (int r = 0; r < 8; ++r) lsum[r] = 0.0f;

#pragma unroll 1
  for (int j0 = 0; j0 < Aseq; j0 += 16) {
    // cooperative stage of K/V j-tiles (16x64 each) into LDS
    for (int u = tid; u < 16 * 16; u += 128) {
      const int row = u >> 4, c4 = u & 15;
      const float4 kv =
          reinterpret_cast<const float4*>(kbase + (size_t)(j0 + row) * HDd)[c4];
      reinterpret_cast<float4*>(&ldsK[row * 68])[c4] = kv;
      const float4 vv =
          reinterpret_cast<const float4*>(vbase + (size_t)(j0 + row) * HDd)[c4];
      reinterpret_cast<float4*>(&ldsV[row * 68])[c4] = vv;
    }
    {  // prefetch next j-tile of the 160MB rel_feat stream
      const int jn = (j0 + 16 < Aseq) ? (j0 + 16) : j0;
      const size_t pidx =
          (((size_t)b * Aseq + i0 + half * 8) * Aseq + jn + nl) * Rr;
      __builtin_prefetch(rel + pidx, 0, 1);
    }
    __syncthreads();
    // S tile = Q[16x64] @ K^T[64x16]
    v8f sacc = {};
#pragma unroll
    for (int s = 0; s < 16; ++s) {
      v2f bf;
      bf.x = ldsK[nl * 68 + 4 * s + kb];
      bf.y = ldsK[nl * 68 + 4 * s + kb + 1];
      sacc = wmma4f32(qa[s], bf, sacc);
    }
    // rel bias + mask + edge weight + exp (C layout: row = r + half*8, col = nl)
    float pr[8];
#pragma unroll
    for (int r = 0; r < 8; ++r) {
      const int i = i0 + r + half * 8;
      const size_t eidx = ((size_t)b * Aseq + i) * Aseq + (j0 + nl);
      const float* rf = rel + eidx * Rr;
      const float bias =
          rf[0] * wr0 + rf[1] * wr1 + rf[2] * wr2 + rf[3] * wr3 + rf[4] * wr4 + brh;
      const float sv = sacc[r] * 0.125f + bias;  // 1/sqrt(64)
      const float p = adj[eidx] ? __expf(sv) * ew[eidx] : 0.0f;
      pr[r] = p;
      lsum[r] += p;
    }
    // C layout -> A layout via per-wave LDS scratch
#pragma unroll
    for (int r = 0; r < 8; ++r) myP[(r + half * 8) * 17 + nl] = pr[r];
    __syncthreads();
    v2f pa[4];
#pragma unroll
    for (int s = 0; s < 4; ++s) {
      pa[s].x = myP[nl * 17 + 4 * s + kb];
      pa[s].y = myP[nl * 17 + 4 * s + kb + 1];
    }
    // O[16x64] += P[16x16] @ V[16x64]  (V from LDS)
#pragma unroll
    for (int s = 0; s < 4; ++s) {
      const int jA = 4 * s + kb;
      v2f bf;
      bf.x = ldsV[jA * 68 + 0 + nl];  bf.y = ldsV[(jA + 1) * 68 + 0 + nl];
      o0 = wmma4f32(pa[s], bf, o0);
      bf.x = ldsV[jA * 68 + 16 + nl]; bf.y = ldsV[(jA + 1) * 68 + 16 + nl];
      o1 = wmma4f32(pa[s], bf, o1);
      bf.x = ldsV[jA * 68 + 32 + nl]; bf.y = ldsV[(jA + 1) * 68 + 32 + nl];
      o2 = wmma4f32(pa[s], bf, o2);
      bf.x = ldsV[jA * 68 + 48 + nl]; bf.y = ldsV[(jA + 1) * 68 + 48 + nl];
      o3 = wmma4f32(pa[s], bf, o3);
    }
    __syncthreads();  // protect ldsK/ldsV/ldsP before next stage
  }
  // reduce per-row partial sums across the 16-lane half
#pragma unroll
  for (int r = 0; r < 8; ++r) {
    float t = lsum[r];
    t += __shfl_xor(t, 1, 32);
    t += __shfl_xor(t, 2, 32);
    t += __shfl_xor(t, 4, 32);
    t += __shfl_xor(t, 8, 32);
    lsum[r] = 1.0f / (t + 1e-8f);
  }
  float* outb = attn_out + (size_t)b * Aseq * Dm;
#pragma unroll
  for (int r = 0; r < 8; ++r) {
    const int i = i0 + r + half * 8;
    const float inv = lsum[r];
    outb[(size_t)i * Dm + h * HDd + 0 + nl]  = o0[r] * inv;
    outb[(size_t)i * Dm + h * HDd + 16 + nl] = o1[r] * inv;
    outb[(size_t)i * Dm + h * HDd + 32 + nl] = o2[r] * inv;
    outb[(size_t)i * Dm + h * HDd + 48 + nl] = o3[r] * inv;
  }
}

// ---------------------------------------------------------------------------
// Kernel 3: generic 8192x256x256 GEMM + bias, EPI: 0 none / 1 relu / 2 +res.
// Block = 4 waves sharing a 32x256 A tile in LDS; 2 m-tiles per wave reuse
// each W fragment load for 2 WMMAs.
// ---------------------------------------------------------------------------
template <int EPI>
__global__ __launch_bounds__(128) void gemm256_kernel(
    const float* __restrict__ Amat, const float* __restrict__ W,
    const float* __restrict__ bias, const float* __restrict__ res,
    float* __restrict__ out) {
  __shared__ float ldsA[32 * 264];
  const int tid = threadIdx.x;
  const int m0 = blockIdx.x * 32;
  for (int u = tid; u < 32 * 64; u += 128) {
    const int row = u >> 6, c4 = u & 63;
    const float4 val =
        reinterpret_cast<const float4*>(Amat + (size_t)(m0 + row) * Dm)[c4];
    reinterpret_cast<float4*>(&ldsA[row * 264])[c4] = val;
  }
  __syncthreads();
  const int lane = tid & 31, wv = tid >> 5;
  const int nl = lane & 15, half = lane >> 4, kb = half * 2;
  const int n = (blockIdx.y * 4 + wv) * 16 + nl;
  const v8f vzero = {};
  v8f acc[2] = {vzero, vzero};
#pragma unroll 4
  for (int s = 0; s < 64; ++s) {
    const int k0 = 4 * s + kb;
    v2f a0, a1, bf;
    a0.x = ldsA[nl * 264 + k0];        a0.y = ldsA[nl * 264 + k0 + 1];
    a1.x = ldsA[(16 + nl) * 264 + k0]; a1.y = ldsA[(16 + nl) * 264 + k0 + 1];
    bf.x = W[k0 * Dm + n];
    bf.y = W[(k0 + 1) * Dm + n];
    acc[0] = wmma4f32(a0, bf, acc[0]);
    acc[1] = wmma4f32(a1, bf, acc[1]);
  }
  const float bn = bias[n];
#pragma unroll
  for (int mt = 0; mt < 2; ++mt) {
#pragma unroll
    for (int r = 0; r < 8; ++r) {
      const int g = m0 + mt * 16 + r + half * 8;
      float vv = acc[mt][r] + bn;
      if constexpr (EPI == 1) vv = vv > 0.0f ? vv : 0.0f;
      if constexpr (EPI == 2) vv += res[(size_t)g * Dm + n];
      out[(size_t)g * Dm + n] = vv;
    }
  }
}

// ---------------------------------------------------------------------------
// Kernel 4: row layernorm over D=256 (one block per row, 8 waves)
// ---------------------------------------------------------------------------
__global__ __launch_bounds__(256) void layernorm_kernel(
    const float* __restrict__ in, const float* __restrict__ g,
    const float* __restrict__ be, float* __restrict__ out) {
  __shared__ float wsum[8];
  const int row = blockIdx.x, t = threadIdx.x;
  const float v = in[(size_t)row * Dm + t];
  float s = v;
#pragma unroll
  for (int m = 1; m < 32; m <<= 1) s += __shfl_xor(s, m, 32);
  if ((t & 31) == 0) wsum[t >> 5] = s;
  __syncthreads();
  float tot = 0.0f;
#pragma unroll
  for (int w = 0; w < 8; ++w) tot += wsum[w];
  const float mean = tot * (1.0f / 256.0f);
  const float d = v - mean;
  __syncthreads();
  float s2 = d * d;
#pragma unroll
  for (int m = 1; m < 32; m <<= 1) s2 += __shfl_xor(s2, m, 32);
  if ((t & 31) == 0) wsum[t >> 5] = s2;
  __syncthreads();
  float tot2 = 0.0f;
#pragma unroll
  for (int w = 0; w < 8; ++w) tot2 += wsum[w];
  const float inv = rsqrtf(tot2 * (1.0f / 256.0f) + 1e-5f);
  out[(size_t)row * Dm + t] = d * inv * g[t] + be[t];
}

extern "C" void kernel_launch(void* const* d_in, const int* in_sizes, int n_in,
                              void* d_out, int out_size, void* d_ws,
                              size_t ws_size, hipStream_t stream) {
  (void)in_sizes; (void)n_in; (void)out_size; (void)ws_size;
  const float* x   = (const float*)d_in[0];
  const float* rel = (const float*)d_in[1];
  const unsigned char* adj = (const unsigned char*)d_in[2];
  const float* ew  = (const float*)d_in[3];
  const float* Wq  = (const float*)d_in[4];  const float* bq  = (const float*)d_in[5];
  const float* Wk  = (const float*)d_in[6];  const float* bk  = (const float*)d_in[7];
  const float* Wv  = (const float*)d_in[8];  const float* bv  = (const float*)d_in[9];
  const float* Wr  = (const float*)d_in[10]; const float* br  = (const float*)d_in[11];
  const float* Wo  = (const float*)d_in[12]; const float* bo  = (const float*)d_in[13];
  const float* W1  = (const float*)d_in[14]; const float* b1  = (const float*)d_in[15];
  const float* W2  = (const float*)d_in[16]; const float* b2  = (const float*)d_in[17];
  const float* g1  = (const float*)d_in[18]; const float* be1 = (const float*)d_in[19];
  const float* g2  = (const float*)d_in[20]; const float* be2 = (const float*)d_in[21];
  float* out = (float*)d_out;
  float* ws  = (float*)d_ws;
  const size_t NE = (size_t)MrowsT * Dm;  // 2M floats per tensor
  float* q  = ws;       float* kk = q + NE;  float* vv = kk + NE;
  float* at = vv + NE;  float* t1 = at + NE; float* h1 = t1 + NE;
  float* t2 = h1 + NE;  float* t3 = t2 + NE;

  qkv_gemm_kernel<<<dim3(MrowsT / 32, 4), 128, 0, stream>>>(
      x, Wq, bq, Wk, bk, Wv, bv, q, kk, vv);
  attn_kernel<<<dim3(Aseq / 64, Hh, Bsz), 128, 0, stream>>>(
      q, kk, vv, rel, adj, ew, Wr, br, at);
  gemm256_kernel<2><<<dim3(MrowsT / 32, 4), 128, 0, stream>>>(at, Wo, bo, x, t1);
  layernorm_kernel<<<MrowsT, 256, 0, stream>>>(t1, g1, be1, h1);
  gemm256_kernel<1><<<dim3(MrowsT / 32, 4), 128, 0, stream>>>(h1, W1, b1,
                                                              nullptr, t2);
  gemm256_kernel<2><<<dim3(MrowsT / 32, 4), 128, 0, stream>>>(t2, W2, b2, h1, t3);
  layernorm_kernel<<<MrowsT, 256, 0, stream>>>(t3, g2, be2, out);
}